// RegionLoss_62964220559940
// MI455X (gfx1250) — compile-verified
//
#include <hip/hip_runtime.h>
#include <hip/hip_bf16.h>
#include <math.h>

// Problem constants (fixed by setup_inputs)
#define NB 128
#define NA 2
#define NH 128
#define NW 128
#define PLANE (NH * NW)          // 16384
#define IGNORE_THRES 0.6f

typedef __attribute__((ext_vector_type(2))) float v2f;
typedef __attribute__((ext_vector_type(8))) float v8f;

// ws layout (floats):
// [0]  noobj_sq_sum (atomic)
// [1]  noobj_count  (atomic)
// [2..9]  Sx, Sy, Sw, Sh, Sconf_obj, R50, R75, SIoU
// [16 + b*8 .. +8) per-image: gx1,gx2,gy1,gy2,a2, best_n(int), gx(int), gy(int)

__device__ __forceinline__ float sigmoid_fast(float x) {
    return 1.0f / (1.0f + __expf(-x));
}

// ---------------- Kernel 1: per-image prep + gathered losses ----------------
__global__ __launch_bounds__(128)
void k_prep(const float* __restrict__ out, const float* __restrict__ tgt,
            const float* __restrict__ anchors, float* __restrict__ ws) {
    __shared__ float red[8][NB];
    int b = threadIdx.x;

    if (b < 2) ws[b] = 0.0f;   // zero atomic accumulators each launch

    float t0 = tgt[b * 4 + 0], t1 = tgt[b * 4 + 1];
    float t2 = tgt[b * 4 + 2], t3 = tgt[b * 4 + 3];
    float gx_f = t0 * (float)NW, gy_f = t1 * (float)NH;
    float gw = t2 * (float)NW, gh = t3 * (float)NH;
    int gx = (int)gx_f, gy = (int)gy_f;

    float a0w = anchors[0], a0h = anchors[1];
    float a1w = anchors[2], a1h = anchors[3];
    float i0 = fminf(gw, a0w) * fminf(gh, a0h);
    float u0 = gw * gh + 1e-16f + a0w * a0h - i0;
    float i1 = fminf(gw, a1w) * fminf(gh, a1h);
    float u1 = gw * gh + 1e-16f + a1w * a1h - i1;
    int best = (i1 / u1 > i0 / u0) ? 1 : 0;       // argmax, first-on-tie
    float aw = best ? a1w : a0w;
    float ah = best ? a1h : a0h;

    // gather predictions at matched cell
    size_t base = ((size_t)b * (NA * 5) + (size_t)best * 5) * PLANE
                + (size_t)gy * NW + gx;
    float ox = out[base + 0 * PLANE];
    float oy = out[base + 1 * PLANE];
    float ow = out[base + 2 * PLANE];
    float oh = out[base + 3 * PLANE];
    float oc = out[base + 4 * PLANE];
    float xo = 1.0f / (1.0f + expf(-ox));
    float yo = 1.0f / (1.0f + expf(-oy));
    float co = 1.0f / (1.0f + expf(-oc));

    float scale = 2.0f - t2 * t3;
    float tx = gx_f - floorf(gx_f);
    float ty = gy_f - floorf(gy_f);
    float tw = logf(gw / aw + 1e-16f);
    float th = logf(gh / ah + 1e-16f);

    // IoU of matched predicted box vs GT (+1 pixel convention)
    float pcx = xo + (float)gx, pcy = yo + (float)gy;
    float pw = expf(ow) * aw, ph = expf(oh) * ah;
    float b1x1 = pcx - pw * 0.5f, b1x2 = pcx + pw * 0.5f;
    float b1y1 = pcy - ph * 0.5f, b1y2 = pcy + ph * 0.5f;
    float g_x1 = gx_f - gw * 0.5f, g_x2 = gx_f + gw * 0.5f;
    float g_y1 = gy_f - gh * 0.5f, g_y2 = gy_f + gh * 0.5f;
    float iw = fmaxf(fminf(b1x2, g_x2) - fmaxf(b1x1, g_x1) + 1.0f, 0.0f);
    float ih = fmaxf(fminf(b1y2, g_y2) - fmaxf(b1y1, g_y1) + 1.0f, 0.0f);
    float inter = iw * ih;
    float area1 = (b1x2 - b1x1 + 1.0f) * (b1y2 - b1y1 + 1.0f);
    float area2 = (g_x2 - g_x1 + 1.0f) * (g_y2 - g_y1 + 1.0f);
    float iou = inter / (area1 + area2 - inter + 1e-16f);

    // per-image params for the streaming kernel
    float* ip = ws + 16 + b * 8;
    ip[0] = g_x1; ip[1] = g_x2; ip[2] = g_y1; ip[3] = g_y2; ip[4] = area2;
    ip[5] = __int_as_float(best);
    ip[6] = __int_as_float(gx);
    ip[7] = __int_as_float(gy);

    float dx = (xo - tx) * scale, dy = (yo - ty) * scale;
    float dw = (ow - tw) * scale, dh = (oh - th) * scale;
    red[0][b] = dx * dx;
    red[1][b] = dy * dy;
    red[2][b] = dw * dw;
    red[3][b] = dh * dh;
    red[4][b] = (co - 1.0f) * (co - 1.0f);
    red[5][b] = (iou > 0.5f) ? 1.0f : 0.0f;
    red[6][b] = (iou > 0.75f) ? 1.0f : 0.0f;
    red[7][b] = iou;
    __syncthreads();
    if (b < 8) {
        float s = 0.0f;
        for (int i = 0; i < NB; ++i) s += red[b][i];
        ws[2 + b] = s;
    }
}

// ---------------- Kernel 2: streaming noobj reduction -----------------------
// 4096 blocks x 256 threads; each block = 1024 contiguous cells of one (b,a)
// plane; each thread does 4 cells via float4 (global_load_b128).
__global__ __launch_bounds__(256)
void k_noobj(const float* __restrict__ out, const float* __restrict__ anchors,
             float* __restrict__ ws) {
    int tid = threadIdx.x;
    int plane_id = blockIdx.x >> 4;   // b*NA + a  (uniform per block)
    int chunk = blockIdx.x & 15;
    int b = plane_id >> 1;
    int a = plane_id & 1;

    const float* ip = ws + 16 + b * 8;
    float g_x1 = ip[0], g_x2 = ip[1], g_y1 = ip[2], g_y2 = ip[3], area2 = ip[4];
    int best = __float_as_int(ip[5]);
    int gx = __float_as_int(ip[6]);
    int gy = __float_as_int(ip[7]);
    float aw = anchors[a * 2 + 0];
    float ah = anchors[a * 2 + 1];
    bool a_match = (a == best);

    int cell0 = chunk * 1024 + tid * 4;
    int h = cell0 >> 7;               // row within plane
    int w0 = cell0 & 127;
    size_t base = (size_t)plane_id * 5 * PLANE + (size_t)cell0;

    float4 o0 = *(const float4*)(out + base + 0 * PLANE);
    float4 o1 = *(const float4*)(out + base + 1 * PLANE);
    float4 o2 = *(const float4*)(out + base + 2 * PLANE);
    float4 o3 = *(const float4*)(out + base + 3 * PLANE);
    float4 o4 = *(const float4*)(out + base + 4 * PLANE);
    const float* p0 = (const float*)&o0;
    const float* p1 = (const float*)&o1;
    const float* p2 = (const float*)&o2;
    const float* p3 = (const float*)&o3;
    const float* p4 = (const float*)&o4;

    float s1 = 0.0f, s2 = 0.0f;
#pragma unroll
    for (int k = 0; k < 4; ++k) {
        float pcx = sigmoid_fast(p0[k]) + (float)(w0 + k);
        float pcy = sigmoid_fast(p1[k]) + (float)h;
        float pw = __expf(p2[k]) * aw;
        float ph = __expf(p3[k]) * ah;
        float conf = sigmoid_fast(p4[k]);

        float b1x1 = pcx - pw * 0.5f, b1x2 = pcx + pw * 0.5f;
        float b1y1 = pcy - ph * 0.5f, b1y2 = pcy + ph * 0.5f;
        float iw = fmaxf(fminf(b1x2, g_x2) - fmaxf(b1x1, g_x1) + 1.0f, 0.0f);
        float ih = fmaxf(fminf(b1y2, g_y2) - fmaxf(b1y1, g_y1) + 1.0f, 0.0f);
        float inter = iw * ih;
        float area1 = (pw + 1.0f) * (ph + 1.0f);
        float iou = inter / (area1 + area2 - inter + 1e-16f);

        bool obj = a_match && (h == gy) && ((w0 + k) == gx);
        bool noobj = (!obj) && !(iou > IGNORE_THRES);
        s1 += noobj ? conf * conf : 0.0f;
        s2 += noobj ? 1.0f : 0.0f;
    }

    // ---- cross-lane reduction via V_WMMA_F32_16X16X4_F32 ----
    // A (16x4, 2 VGPRs/lane): A[m][0]=s1_m, A[m][1]=s2_m,
    //                         A[m][2]=s1_{m+16}, A[m][3]=s2_{m+16}
    // B selector: col0=(1,0,1,0)^T, col1=(0,1,0,1)^T
    //   B VGPR0: lanes0-15=K0 row, lanes16-31=K1; VGPR1: K2/K3
    //   -> b.x=b.y=1 only on lanes 0 and 17.
    // D[m][0] = s1_m + s1_{m+16}; D[m][1] = s2_m + s2_{m+16}
    int lane = tid & 31;
    int wave = tid >> 5;
    v2f A; A.x = s1; A.y = s2;
    float bs = (lane == 0 || lane == 17) ? 1.0f : 0.0f;
    v2f B; B.x = bs; B.y = bs;
    v8f C = {0.f, 0.f, 0.f, 0.f, 0.f, 0.f, 0.f, 0.f};
    v8f D = __builtin_amdgcn_wmma_f32_16x16x4_f32(
        false, A, false, B, (short)0, C, false, false);
    // lane 0: rows 0..7 of col 0; lane 16: rows 8..15 of col 0 (s1)
    // lane 1 / lane 17: same for col 1 (s2)
    float t = D[0] + D[1] + D[2] + D[3] + D[4] + D[5] + D[6] + D[7];
    t += __shfl_down(t, 16, 32);   // lane0 -> wave s1, lane1 -> wave s2

    __shared__ float lds1[8], lds2[8];
    if (lane == 0) lds1[wave] = t;
    if (lane == 1) lds2[wave] = t;
    __syncthreads();
    if (tid == 0) {
        float s = 0.0f;
        for (int i = 0; i < 8; ++i) s += lds1[i];
        atomicAdd(&ws[0], s);
    }
    if (tid == 1) {
        float s = 0.0f;
        for (int i = 0; i < 8; ++i) s += lds2[i];
        atomicAdd(&ws[1], s);
    }
}

// ---------------- Kernel 3: combine scalars ---------------------------------
__global__ void k_final(const float* __restrict__ ws, float* __restrict__ outp) {
    const float inv = 1.0f / (float)NB;
    float noobj_term = ws[0] / fmaxf(ws[1], 1.0f);
    float loss = (ws[2] + ws[3] + ws[4] + ws[5]) * inv
               + 5.0f * ws[6] * inv + 1.0f * noobj_term;
    outp[0] = loss;
    outp[1] = ws[7] * inv;   // recall50
    outp[2] = ws[8] * inv;   // recall75
    outp[3] = ws[9] * inv;   // avg_iou
}

extern "C" void kernel_launch(void* const* d_in, const int* in_sizes, int n_in,
                              void* d_out, int out_size, void* d_ws, size_t ws_size,
                              hipStream_t stream) {
    const float* output  = (const float*)d_in[0];   // [128, 10, 128, 128]
    const float* targets = (const float*)d_in[1];   // [128, 4]
    const float* anchors = (const float*)d_in[2];   // [2, 2]
    float* ws = (float*)d_ws;
    float* out = (float*)d_out;                     // 4 scalars

    k_prep<<<1, 128, 0, stream>>>(output, targets, anchors, ws);
    k_noobj<<<NB * NA * 16, 256, 0, stream>>>(output, anchors, ws);
    k_final<<<1, 1, 0, stream>>>(ws, out);
}